// DiceCEsoft_9423158247527
// MI455X (gfx1250) — compile-verified
//
#include <hip/hip_runtime.h>

// CDNA5 / gfx1250: wave32, WMMA f32 16x16x4.
typedef float v2f __attribute__((ext_vector_type(2)));
typedef float v4f __attribute__((ext_vector_type(4)));
typedef float v8f __attribute__((ext_vector_type(8)));
typedef int   v4i __attribute__((ext_vector_type(4)));

#define HWD (128 * 128 * 128)   // 2,097,152 spatial sites
#define EPSF 1e-10f
#define SMOOTHF 1e-5f
#define LN2F 0.6931471805599453f

// pred viewed as 16 x HWD matrix: row r = b*8+n, pred[r*HWD + site].
// Per wave-iteration: 8 consecutive sites, split into two 16x4 WMMA tiles
// (tileA: sites {0,1,4,5}, tileB: {2,3,6,7} -- K-permutation invariant).
// A layout (ISA 7.12.2): lane m holds K=0,1 ; lane m+16 holds K=2,3.
// B layout (C/D convention): VGPR0 = rows K0 (lanes0-15) / K2 (lanes16-31),
//                            VGPR1 = rows K1 / K3; column c = lane&15.
// CE matrix uses log2 (raw v_log_f32); ln2 factor applied once at finalize.
__global__ __launch_bounds__(256) void dicece_main(
    const float* __restrict__ pred, const int* __restrict__ target,
    float* __restrict__ ws, int sitesPerWave)
{
    __shared__ float sInter[8 * 256];   // [wave][lane][vgpr]
    __shared__ float sCe[8 * 256];
    __shared__ float sPo[8 * 32];
    __shared__ float sG[8 * 32];
    __shared__ float sCeR[16];

    const int tid  = threadIdx.x;
    const int w    = tid >> 5;          // wave index in block (8 waves)
    const int lane = tid & 31;
    const int lo   = lane & 15;
    const int hi   = lane >> 4;         // 0: K0/K1 half, 1: K2/K3 half

    const long gwave     = (long)blockIdx.x * 8 + w;
    const long siteBase0 = gwave * (long)sitesPerWave;

    const float* prow = pred + (long)lo * HWD + siteBase0 + hi * 4; // A row m = lo
    const int    ncol = lo & 7;                                     // B col (b',n')
    const int*   trow = target + (long)(lo >> 3) * HWD + siteBase0 + hi * 4;

    v8f accI = {};   // D-accumulator: diagonal = inter[b,n]
    v8f accC = {};   // D-accumulator: CE contractions (log2 domain)
    float po = 0.f, g = 0.f;

    const int iters = sitesPerWave >> 3;
#pragma unroll 4
    for (int i = 0; i < iters; ++i) {
        // Streamed exactly once: non-temporal to keep L2 clean.
        const v4f p  = __builtin_nontemporal_load((const v4f*)(prow + (long)i * 8));
        const v4i tt = __builtin_nontemporal_load((const v4i*)(trow + (long)i * 8));

        // tileA A-operand = sites {0,1 | 4,5}, tileB = sites {2,3 | 6,7}
        v2f aA = { p.x, p.y };
        v2f aB = { p.z, p.w };
        // raw v_log_f32 (log2); inputs >= 1e-10 so no denorm fixup needed
        v2f lA = { __builtin_amdgcn_logf(p.x + EPSF),
                   __builtin_amdgcn_logf(p.y + EPSF) };
        v2f lB = { __builtin_amdgcn_logf(p.z + EPSF),
                   __builtin_amdgcn_logf(p.w + EPSF) };
        // one-hot B operands, matching per-K site mapping
        v2f bA = { (tt.x == ncol) ? 1.f : 0.f, (tt.y == ncol) ? 1.f : 0.f };
        v2f bB = { (tt.z == ncol) ? 1.f : 0.f, (tt.w == ncol) ? 1.f : 0.f };

        accI = __builtin_amdgcn_wmma_f32_16x16x4_f32(false, aA, false, bA,
                                                     (short)0, accI, false, false);
        accI = __builtin_amdgcn_wmma_f32_16x16x4_f32(false, aB, false, bB,
                                                     (short)0, accI, false, false);
        accC = __builtin_amdgcn_wmma_f32_16x16x4_f32(false, lA, false, bA,
                                                     (short)0, accC, false, false);
        accC = __builtin_amdgcn_wmma_f32_16x16x4_f32(false, lB, false, bB,
                                                     (short)0, accC, false, false);

        po += p.x + p.y + p.z + p.w;          // row-partial of pred_o
        g  += bA.x + bA.y + bB.x + bB.y;      // column-partial of ground_o
    }

    // Stage D accumulators to LDS: [lane][vgpr j] so (M,N) -> addr is direct.
    float* si = sInter + w * 256 + lane * 8;
    float* sc = sCe    + w * 256 + lane * 8;
#pragma unroll
    for (int j = 0; j < 8; ++j) { si[j] = accI[j]; sc[j] = accC[j]; }
    sPo[w * 32 + lane] = po;
    sG[w * 32 + lane]  = g;
    __syncthreads();

    // Extract needed D entries and reduce the block's 8 waves in fixed order.
    // C/D layout: VGPR j: lanes0-15 -> (M=j, N=lane); lanes16-31 -> (M=j+8, N=lane-16)
    if (tid < 16) {
        const int r = tid;             // (b,n) = (r>>3, r&7)
        const int n = r & 7;
        const int j = r & 7;
        const int laneD  = (r < 8) ? r        : r + 16;   // diagonal (M=r,N=r)
        const int laneC1 = (r < 8) ? n        : n + 16;   // (M=r, N=(0,n))
        const int laneC2 = (r < 8) ? (8 + n)  : (24 + n); // (M=r, N=(1,n))
        float interR = 0.f, ceR = 0.f, poR = 0.f, gR = 0.f;
        for (int ww = 0; ww < 8; ++ww) {
            interR += sInter[ww * 256 + laneD * 8 + j];
            ceR    += sCe[ww * 256 + laneC1 * 8 + j]
                    + sCe[ww * 256 + laneC2 * 8 + j];
            poR    += sPo[ww * 32 + r] + sPo[ww * 32 + r + 16];
            gR     += sG[ww * 32 + r] + sG[ww * 32 + r + 16];
        }
        float* outp = ws + (long)blockIdx.x * 64;
        outp[1 + r]  = interR;
        outp[17 + r] = poR;
        outp[33 + r] = gR;
        sCeR[r] = ceR;
    }
    __syncthreads();
    if (tid == 0) {
        float c = 0.f;
        for (int r2 = 0; r2 < 16; ++r2) c += sCeR[r2];
        ws[(long)blockIdx.x * 64] = c;
    }
}

// Deterministic final reduction over per-block partials, then scalar combine.
__global__ __launch_bounds__(512) void dicece_final(
    const float* __restrict__ ws, float* __restrict__ out, int nblocks)
{
    __shared__ float sPart[49 * 8];
    __shared__ float sAcc[49];
    const int tid = threadIdx.x;
    const int j = tid >> 3;          // accumulator id 0..63 (49 used)
    const int c = tid & 7;           // chunk id
    const int chunk = nblocks >> 3;
    if (j < 49) {
        float a = 0.f;
        const int b0 = c * chunk;
        for (int b = 0; b < chunk; ++b)
            a += ws[(long)(b0 + b) * 64 + j];
        sPart[j * 8 + c] = a;
    }
    __syncthreads();
    if (tid < 49) {
        float a = 0.f;
        for (int cc = 0; cc < 8; ++cc) a += sPart[tid * 8 + cc];
        sAcc[tid] = a;
    }
    __syncthreads();
    if (tid == 0) {
        const float ceT = sAcc[0] * LN2F;   // log2 -> ln, applied once
        float dice = 0.f;
        for (int r = 0; r < 16; ++r) {
            const float inter = sAcc[1 + r];
            const float poV   = sAcc[17 + r];
            const float gV    = sAcc[33 + r];
            dice += 1.f - (2.f * inter + SMOOTHF) / (gV + poV + SMOOTHF);
        }
        dice *= (1.f / 16.f);
        const float ce = -ceT / (2.f * 2.f * (float)HWD); // /(B*HWD) then /B
        out[0] = ce + dice;
    }
}

extern "C" void kernel_launch(void* const* d_in, const int* in_sizes, int n_in,
                              void* d_out, int out_size, void* d_ws, size_t ws_size,
                              hipStream_t stream) {
    const float* pred   = (const float*)d_in[0];
    const int*   target = (const int*)d_in[1];
    float* ws = (float*)d_ws;

    // 1024 blocks x 256 threads = 8192 waves; each wave owns a contiguous
    // 256-site chunk (exactly divides HWD). Shrink if workspace is tight.
    int nblocks = 1024;
    while ((size_t)nblocks * 64 * sizeof(float) > ws_size && nblocks > 64)
        nblocks >>= 1;
    const int sitesPerWave = HWD / (nblocks * 8);

    dicece_main<<<nblocks, 256, 0, stream>>>(pred, target, ws, sitesPerWave);
    dicece_final<<<1, 512, 0, stream>>>(ws, (float*)d_out, nblocks);
}